// GMLayer_37331855736965
// MI455X (gfx1250) — compile-verified
//
#include <hip/hip_runtime.h>
#include <math.h>

// ---------------------------------------------------------------------------
// GMM layer for MI455X (gfx1250, wave32, WMMA f32 path)
//   out[b,s,g] = exp(-0.5 * ( (x-c)^T Ci (x-c) + log((2pi)^F det) ))
// Fused GEMM: acc[n,g] = sum_k phi[n,k] * Bmat[k,g]
//   K = 1024 (outer products x_i*x_j) + 32 (linear, -2*Ci*c) + 1 (const) + 3 pad
// ---------------------------------------------------------------------------

typedef __attribute__((ext_vector_type(2))) float v2f;
typedef __attribute__((ext_vector_type(8))) float v8f;

#define B_SZ 8
#define S_SZ 4096
#define G_SZ 128
#define F_SZ 32
#define NPTS (B_SZ * S_SZ)       // 32768
#define NTILES (NPTS / 16)       // 2048
#define NCHUNK_MAIN 256          // 1024 quadratic K-values / 4
#define NCHUNK 265               // + 9 tail chunks (36 K-values)
#define ROWSTRIDE 36             // conflict-free LDS stride for x_i broadcast
#define WPB 8                    // waves per block (main kernel)

// B-buffer layout (in d_ws): float Bbuf[NCHUNK][32 lanes][16 floats]
//   lane L, floats [2*nt, 2*nt+1] = B-fragment VGPR0/VGPR1 for N-tile nt.
//   WMMA 16x16x4 f32 layout: lanes 0-15 hold K rows (jb+0, jb+1),
//   lanes 16-31 hold K rows (jb+2, jb+3); N = lane % 16.

__device__ __forceinline__ v8f wmma4(v2f a, float bx, float by, v8f c) {
    v2f b; b.x = bx; b.y = by;
    return __builtin_amdgcn_wmma_f32_16x16x4_f32(
        /*neg_a=*/false, a, /*neg_b=*/false, b,
        /*c_mod=*/(short)0, c, /*reuse_a=*/false, /*reuse_b=*/false);
}

// ---------------------------------------------------------------------------
// Prep: invert each 32x32 covariance (Gauss-Jordan, SPD -> diagonal pivots),
// compute log-det, v = Ci*c, constant term; scatter into WMMA-B layout.
// One wave per gaussian.
// ---------------------------------------------------------------------------
__device__ __forceinline__ float tailval(int k, const float* vsh, float w) {
    if (k < 32) return -2.0f * vsh[k];
    if (k == 32) return w;
    return 0.0f;
}

__global__ __launch_bounds__(32) void gm_prep(const float* __restrict__ centers,
                                              const float* __restrict__ cov,
                                              float* __restrict__ bbuf) {
    __shared__ float A[32 * 33];
    __shared__ float Inv[32 * 33];
    __shared__ float tmpcol[32];
    __shared__ float cs[32];
    __shared__ float vsh[32];

    const int g = blockIdx.x;
    const int j = threadIdx.x;          // column owned by this lane

    for (int i = 0; i < 32; ++i) {
        A[i * 33 + j]   = cov[(size_t)g * 1024 + i * 32 + j];
        Inv[i * 33 + j] = (i == j) ? 1.0f : 0.0f;
    }
    cs[j] = centers[g * 32 + j];
    __syncthreads();

    float logdet = 0.0f;
    for (int k = 0; k < 32; ++k) {
        const float piv = A[k * 33 + k];
        logdet += logf(piv);
        const float ip = 1.0f / piv;
        A[k * 33 + j]   *= ip;          // scale pivot row (own column only)
        Inv[k * 33 + j] *= ip;
        tmpcol[j] = A[j * 33 + k];      // snapshot pivot column
        __syncthreads();
        for (int i = 0; i < 32; ++i) {
            if (i == k) continue;
            const float f = tmpcol[i];
            A[i * 33 + j]   -= f * A[k * 33 + j];
            Inv[i * 33 + j] -= f * Inv[k * 33 + j];
        }
        __syncthreads();
    }

    // v = Ci * c (lane j computes row j), then c.v
    float vv = 0.0f;
    for (int t = 0; t < 32; ++t) vv += Inv[j * 33 + t] * cs[t];
    vsh[j] = vv;
    __syncthreads();
    float dot = 0.0f;
    for (int t = 0; t < 32; ++t) dot += cs[t] * vsh[t];
    const float w = dot + 32.0f * 1.8378770664093453f + logdet; // + F*log(2pi)

    const int ll = g & 15;              // lane slot (N = lane%16)
    const int nt = g >> 4;              // N-tile

    // quadratic chunks: chunk c covers K rows (i, jb..jb+3), phi = x_i * x_j
    for (int c = j; c < NCHUNK_MAIN; c += 32) {
        const int i  = c >> 3;
        const int jb = (c & 7) << 2;
        const size_t base = (size_t)c * 512;
        bbuf[base + (size_t)ll * 16 + 2 * nt + 0]        = Inv[i * 33 + jb + 0];
        bbuf[base + (size_t)ll * 16 + 2 * nt + 1]        = Inv[i * 33 + jb + 1];
        bbuf[base + (size_t)(ll + 16) * 16 + 2 * nt + 0] = Inv[i * 33 + jb + 2];
        bbuf[base + (size_t)(ll + 16) * 16 + 2 * nt + 1] = Inv[i * 33 + jb + 3];
    }
    // tail chunks: phi = x_j (j<32), 1 (slot 32), 0 (pad)
    for (int c = NCHUNK_MAIN + j; c < NCHUNK; c += 32) {
        const int jb = (c - NCHUNK_MAIN) << 2;
        const size_t base = (size_t)c * 512;
        bbuf[base + (size_t)ll * 16 + 2 * nt + 0]        = tailval(jb + 0, vsh, w);
        bbuf[base + (size_t)ll * 16 + 2 * nt + 1]        = tailval(jb + 1, vsh, w);
        bbuf[base + (size_t)(ll + 16) * 16 + 2 * nt + 0] = tailval(jb + 2, vsh, w);
        bbuf[base + (size_t)(ll + 16) * 16 + 2 * nt + 1] = tailval(jb + 3, vsh, w);
    }
}

// ---------------------------------------------------------------------------
// Main: one wave per 16-point tile; 8 f32-WMMA accumulators cover all G=128.
// x_j operands in registers; LDS only broadcasts x_i (1 ds_load / 8 chunks).
// B fragments are double-buffered: next chunk's 4x b128 loads are in flight
// while the current octet of WMMAs executes.
// ---------------------------------------------------------------------------
#define WMMA_OCTET4(AV, B0, B1, B2, B3)                           \
    {                                                             \
        acc[0] = wmma4(AV, (B0).x, (B0).y, acc[0]);               \
        acc[1] = wmma4(AV, (B0).z, (B0).w, acc[1]);               \
        acc[2] = wmma4(AV, (B1).x, (B1).y, acc[2]);               \
        acc[3] = wmma4(AV, (B1).z, (B1).w, acc[3]);               \
        acc[4] = wmma4(AV, (B2).x, (B2).y, acc[4]);               \
        acc[5] = wmma4(AV, (B2).z, (B2).w, acc[5]);               \
        acc[6] = wmma4(AV, (B3).x, (B3).y, acc[6]);               \
        acc[7] = wmma4(AV, (B3).z, (B3).w, acc[7]);               \
    }

__global__ __launch_bounds__(32 * WPB) void gm_main(const float* __restrict__ x,
                                                    const float* __restrict__ bbuf,
                                                    float* __restrict__ out) {
    __shared__ float lx[WPB * 16 * ROWSTRIDE];

    const int lane = threadIdx.x & 31;
    const int wave = threadIdx.x >> 5;
    const int tile = blockIdx.x * WPB + wave;
    const int row  = lane & 15;
    const int hi   = lane >> 4;         // 0: K rows +0/+1, 1: K rows +2/+3
    float* myx = &lx[wave * 16 * ROWSTRIDE];

    // lane's own point row -> registers (both half-waves load the same row)
    float4 r[8];
    if (tile < NTILES) {
        const float4* xsrc =
            reinterpret_cast<const float4*>(x + (size_t)(tile * 16 + row) * 32);
#pragma unroll
        for (int q = 0; q < 8; ++q) r[q] = xsrc[q];
        // stage row into LDS for the x_i broadcast path (lanes 0-15 only)
        if (hi == 0) {
            float4* dst = reinterpret_cast<float4*>(&myx[row * ROWSTRIDE]);
#pragma unroll
            for (int q = 0; q < 8; ++q) dst[q] = r[q];
        }
    }
    __syncthreads();
    if (tile >= NTILES) return;

    // per-lane invariant x_j operands: xj[jc] = x[row][4*jc + 2*hi + {0,1}]
    float xjA[8], xjB[8];
#pragma unroll
    for (int q = 0; q < 8; ++q) {
        xjA[q] = hi ? r[q].z : r[q].x;
        xjB[q] = hi ? r[q].w : r[q].y;
    }

    v8f acc[8] = {};
    const float4* bl = reinterpret_cast<const float4*>(bbuf) + lane * 4;
    const float* xr = &myx[row * ROWSTRIDE];

    // preload chunk 0's B fragments
    float4 c0 = bl[0], c1 = bl[1], c2 = bl[2], c3 = bl[3];

    // quadratic part: K-chunk c = i*8 + jc, phi = x_i * x_{4*jc+..}
    for (int i = 0; i < 32; ++i) {
        const float xi = xr[i];                 // one ds_load_b32 per 8 chunks
        const float4* bpi = bl + (size_t)i * 1024;   // chunk i*8
#pragma unroll
        for (int jc = 0; jc < 8; ++jc) {
            // prefetch next chunk (at jc==7 this crosses into chunk (i+1)*8,
            // and at i==31 into tail chunk 256 -- the buffer is linear)
            const float4* np = bpi + (size_t)(jc + 1) * 128;
            const float4 n0 = np[0], n1 = np[1], n2 = np[2], n3 = np[3];
            v2f a; a.x = xi * xjA[jc]; a.y = xi * xjB[jc];
            WMMA_OCTET4(a, c0, c1, c2, c3);
            c0 = n0; c1 = n1; c2 = n2; c3 = n3;
        }
    }

    // linear tail: phi = x_j (chunks 256..263), then constant chunk 264
    {
        const float4* bpt = bl + (size_t)NCHUNK_MAIN * 128;
#pragma unroll
        for (int t = 0; t < 8; ++t) {
            const float4* np = bpt + (size_t)(t + 1) * 128;   // up to chunk 264
            const float4 n0 = np[0], n1 = np[1], n2 = np[2], n3 = np[3];
            v2f a; a.x = xjA[t]; a.y = xjB[t];
            WMMA_OCTET4(a, c0, c1, c2, c3);
            c0 = n0; c1 = n1; c2 = n2; c3 = n3;
        }
        // constant chunk: K slots 32..35 -> phi = {1,0,0,0}; no prefetch
        v2f a; a.x = hi ? 0.0f : 1.0f; a.y = 0.0f;
        WMMA_OCTET4(a, c0, c1, c2, c3);
    }

    // epilogue: C/D layout -> VGPR r: lanes0-15 M=r, lanes16-31 M=8+r; N=lane%16
    const size_t nbase = (size_t)tile * 16;
#pragma unroll
    for (int nt = 0; nt < 8; ++nt) {
#pragma unroll
        for (int rr = 0; rr < 8; ++rr) {
            const int M = rr + hi * 8;
            out[(nbase + M) * G_SZ + nt * 16 + row] = __expf(-0.5f * acc[nt][rr]);
        }
    }
}

extern "C" void kernel_launch(void* const* d_in, const int* in_sizes, int n_in,
                              void* d_out, int out_size, void* d_ws, size_t ws_size,
                              hipStream_t stream) {
    const float* x       = (const float*)d_in[0];
    const float* centers = (const float*)d_in[1];
    const float* cov     = (const float*)d_in[2];
    float* out  = (float*)d_out;
    float* bbuf = (float*)d_ws;   // needs 542,720 bytes

    gm_prep<<<G_SZ, 32, 0, stream>>>(centers, cov, bbuf);
    gm_main<<<NTILES / WPB, 32 * WPB, 0, stream>>>(x, bbuf, out);
}